// AttnDecoderRNN_27831388078905
// MI455X (gfx1250) — compile-verified
//
#include <hip/hip_runtime.h>
#include <math.h>

#define Hdim 100
#define Vdim 19950
#define Bdim 128
#define Sdim 50
#define Tdim 50
#define BOS_TOK 1

typedef __attribute__((ext_vector_type(2))) float v2f;
typedef __attribute__((ext_vector_type(8))) float v8f;

// ---------------------------------------------------------------------------
// Generic f32 WMMA GEMM (tile-per-wave):
//   out[m, n] = sum_k A[m,k] * W[n,k] + bias[n]
// Used once for keys_proj (M=6400, N=K=100).
// ---------------------------------------------------------------------------
__global__ void gemm_wmma_f32(const float* __restrict__ A, int lda,
                              const float* __restrict__ W, int ldw,
                              const float* __restrict__ bias,
                              float* __restrict__ out, long long ldc,
                              int N, int K, int Ntiles, int totalTiles)
{
    const int wave = threadIdx.x >> 5;
    const int tile = blockIdx.x * (blockDim.x >> 5) + wave;
    if (tile >= totalTiles) return;            // uniform per wave

    const int mtile = tile / Ntiles;
    const int ntile = tile % Ntiles;
    const int lane  = threadIdx.x & 31;
    const int half  = lane >> 4;               // selects K pair (A/B frags)
    const int l16   = lane & 15;

    const int  arow   = mtile * 16 + l16;
    const int  bcol   = ntile * 16 + l16;
    const bool bvalid = (bcol < N);

    const float* __restrict__ Ap = A + (long long)arow * lda + half * 2;
    const float* __restrict__ Wp = W + (long long)(bvalid ? bcol : 0) * ldw + half * 2;

    v8f c = {};
    for (int kk = 0; kk < K; kk += 4) {
        v2f a, b;
        a.x = Ap[kk];
        a.y = Ap[kk + 1];
        float b0 = Wp[kk];
        float b1 = Wp[kk + 1];
        b.x = bvalid ? b0 : 0.0f;
        b.y = bvalid ? b1 : 0.0f;
        c = __builtin_amdgcn_wmma_f32_16x16x4_f32(
                false, a, false, b, (short)0, c, false, false);
    }

    if (bvalid) {
        const float bb = bias[bcol];
        #pragma unroll
        for (int r = 0; r < 8; ++r) {
            const int row = mtile * 16 + half * 8 + r;   // C/D: M = r + 8*half
            out[(long long)row * ldc + bcol] = c[r] + bb;
        }
    }
}

// ---------------------------------------------------------------------------
// Logits GEMM, M blocked in registers: each wave owns a full 128-row column
// stripe (8 M-tiles x one N-tile). Every W_out fragment is loaded ONCE and
// fed to 8 independent V_WMMA_F32_16X16X4_F32 ops (hides WMMA RAW hazards,
// cuts L2 traffic on the 8 MB W_out operand by 8x).
//   out[m, n] = sum_k A[m,k] * W[n,k] + bias[n],  A: 128 x 100, W: V x 100
// ---------------------------------------------------------------------------
__global__ void logits_wmma_f32(const float* __restrict__ A,    // h1 (B x H)
                                const float* __restrict__ W,    // W_out (V x H)
                                const float* __restrict__ bias, // b_out (V)
                                float* __restrict__ out,        // logp + t*V
                                long long ldc)                  // T*V
{
    const int NT = (Vdim + 15) / 16;           // 1247 column tiles
    const int wave  = threadIdx.x >> 5;
    const int ntile = blockIdx.x * (blockDim.x >> 5) + wave;
    if (ntile >= NT) return;                   // uniform per wave

    const int lane = threadIdx.x & 31;
    const int half = lane >> 4;
    const int l16  = lane & 15;

    const int  bcol   = ntile * 16 + l16;
    const bool bvalid = (bcol < Vdim);

    const float* __restrict__ Wp = W + (long long)(bvalid ? bcol : 0) * Hdim + half * 2;
    const float* __restrict__ Ap = A + (long long)l16 * Hdim + half * 2;

    v8f acc[8];
    #pragma unroll
    for (int m = 0; m < 8; ++m) acc[m] = (v8f){};

    for (int kk = 0; kk < Hdim; kk += 4) {
        v2f b;
        float b0 = Wp[kk];
        float b1 = Wp[kk + 1];
        b.x = bvalid ? b0 : 0.0f;
        b.y = bvalid ? b1 : 0.0f;
        #pragma unroll
        for (int m = 0; m < 8; ++m) {          // 8 independent WMMAs per B-frag
            v2f a;
            a.x = Ap[m * 16 * Hdim + kk];
            a.y = Ap[m * 16 * Hdim + kk + 1];
            acc[m] = __builtin_amdgcn_wmma_f32_16x16x4_f32(
                         false, a, false, b, (short)0, acc[m], false, false);
        }
    }

    if (bvalid) {
        const float bb = bias[bcol];
        #pragma unroll
        for (int m = 0; m < 8; ++m) {
            #pragma unroll
            for (int r = 0; r < 8; ++r) {
                const int row = m * 16 + half * 8 + r;
                out[(long long)row * ldc + bcol] = acc[m][r] + bb;
            }
        }
    }
}

// ---------------------------------------------------------------------------
// Per-step fused attention + LSTM cell. One block per batch row b.
// ---------------------------------------------------------------------------
__global__ void step_kernel(const float* __restrict__ eo,    // (B,S,H)
                            const float* __restrict__ keys,  // (B,S,H) = Ua proj
                            const float* __restrict__ c0,    // (B,H) slice of c0_seq
                            const int*   __restrict__ tgt,   // (B,T)
                            const float* __restrict__ emb,   // (V,H)
                            const float* __restrict__ Wa, const float* __restrict__ ba,
                            const float* __restrict__ Va, const float* __restrict__ bv,
                            const float* __restrict__ Wih, const float* __restrict__ bih,
                            const float* __restrict__ Whh, const float* __restrict__ bhh,
                            const float* __restrict__ hin, float* __restrict__ hout,
                            float* __restrict__ attn,        // attn base + t*S
                            int t)
{
    __shared__ float sh[Hdim];
    __shared__ float sq[Hdim];
    __shared__ float sw[Sdim];
    __shared__ float sx[2 * Hdim];
    __shared__ float sg[4 * Hdim];

    const int b = blockIdx.x;
    const int tid = threadIdx.x;

    if (tid < Hdim) sh[tid] = hin[b * Hdim + tid];
    __syncthreads();

    if (tid < Hdim) {                       // q = h * Wa^T + ba
        float acc = ba[tid];
        const float* wr = Wa + (long long)tid * Hdim;
        for (int k = 0; k < Hdim; ++k) acc += sh[k] * wr[k];
        sq[tid] = acc;
    }
    __syncthreads();

    if (tid < Sdim) {                       // scores
        const float* kp = keys + ((long long)b * Sdim + tid) * Hdim;
        float acc = bv[0];
        for (int k = 0; k < Hdim; ++k) acc += tanhf(sq[k] + kp[k]) * Va[k];
        sw[tid] = acc;
    }
    __syncthreads();

    if (tid == 0) {                         // softmax over S=50 (tiny, serial)
        float m = sw[0];
        for (int s = 1; s < Sdim; ++s) m = fmaxf(m, sw[s]);
        float sum = 0.0f;
        for (int s = 0; s < Sdim; ++s) { float e = expf(sw[s] - m); sw[s] = e; sum += e; }
        float inv = 1.0f / sum;
        for (int s = 0; s < Sdim; ++s) sw[s] *= inv;
    }
    __syncthreads();

    if (tid < Sdim)                         // attentions output (B,T,S)
        attn[(long long)b * Tdim * Sdim + tid] = sw[tid];

    if (tid < Hdim) {                       // context + embedding -> x = [emb||ctx]
        float acc = 0.0f;
        const float* eb = eo + (long long)b * Sdim * Hdim + tid;
        for (int s = 0; s < Sdim; ++s) acc += sw[s] * eb[(long long)s * Hdim];
        sx[Hdim + tid] = acc;
        const int tok = (t == 0) ? BOS_TOK : tgt[b * Tdim + (t - 1)];
        sx[tid] = emb[(long long)tok * Hdim + tid];
    }
    __syncthreads();

    for (int j = tid; j < 4 * Hdim; j += blockDim.x) {   // gates
        float acc = bih[j] + bhh[j];
        const float* wi = Wih + (long long)j * (2 * Hdim);
        for (int k = 0; k < 2 * Hdim; ++k) acc += sx[k] * wi[k];
        const float* wh = Whh + (long long)j * Hdim;
        for (int k = 0; k < Hdim; ++k) acc += sh[k] * wh[k];
        sg[j] = acc;
    }
    __syncthreads();

    if (tid < Hdim) {                       // LSTM cell elementwise
        const float gi = sg[tid];
        const float gf = sg[Hdim + tid];
        const float gg = sg[2 * Hdim + tid];
        const float go = sg[3 * Hdim + tid];
        const float c0v = c0[b * Hdim + tid];
        const float si = 1.0f / (1.0f + expf(-gi));
        const float sf = 1.0f / (1.0f + expf(-gf));
        const float so = 1.0f / (1.0f + expf(-go));
        const float c1 = sf * c0v + si * tanhf(gg);
        hout[b * Hdim + tid] = so * tanhf(c1);
    }
}

// ---------------------------------------------------------------------------
// In-place log_softmax over one row of V=19950 logits (one block per batch b).
// ---------------------------------------------------------------------------
__global__ void logsoftmax_kernel(float* __restrict__ base, long long ldc)
{
    __shared__ float red[256];
    const int b = blockIdx.x;
    const int tid = threadIdx.x;
    float* row = base + (long long)b * ldc;

    float m = -3.0e38f;
    for (int v = tid; v < Vdim; v += blockDim.x) m = fmaxf(m, row[v]);
    red[tid] = m; __syncthreads();
    for (int s = 128; s > 0; s >>= 1) {
        if (tid < s) red[tid] = fmaxf(red[tid], red[tid + s]);
        __syncthreads();
    }
    const float M = red[0];
    __syncthreads();

    float acc = 0.0f;
    for (int v = tid; v < Vdim; v += blockDim.x) acc += expf(row[v] - M);
    red[tid] = acc; __syncthreads();
    for (int s = 128; s > 0; s >>= 1) {
        if (tid < s) red[tid] += red[tid + s];
        __syncthreads();
    }
    const float lse = M + logf(red[0]);
    __syncthreads();

    for (int v = tid; v < Vdim; v += blockDim.x) row[v] -= lse;
}

__global__ void copy_kernel(float* __restrict__ dst, const float* __restrict__ src, int n)
{
    const int i = blockIdx.x * blockDim.x + threadIdx.x;
    if (i < n) dst[i] = src[i];
}

// ---------------------------------------------------------------------------
extern "C" void kernel_launch(void* const* d_in, const int* in_sizes, int n_in,
                              void* d_out, int out_size, void* d_ws, size_t ws_size,
                              hipStream_t stream)
{
    const float* eo   = (const float*)d_in[0];   // encoder_outputs (B,S,H)
    const float* ehid = (const float*)d_in[1];   // encoder_hidden (1,B,H)
    const float* c0s  = (const float*)d_in[2];   // c0_seq (T,B,H)
    const int*   tgt  = (const int*)  d_in[3];   // target_tensor (B,T) int32
    const float* emb  = (const float*)d_in[4];   // embedding (V,H)
    const float* Wa   = (const float*)d_in[5];
    const float* ba   = (const float*)d_in[6];
    const float* Ua   = (const float*)d_in[7];
    const float* bu   = (const float*)d_in[8];
    const float* Va   = (const float*)d_in[9];
    const float* bv   = (const float*)d_in[10];
    const float* Wih  = (const float*)d_in[11];
    const float* bih  = (const float*)d_in[12];
    const float* Whh  = (const float*)d_in[13];
    const float* bhh  = (const float*)d_in[14];
    const float* Wout = (const float*)d_in[15];
    const float* bout = (const float*)d_in[16];

    float* out  = (float*)d_out;
    float* logp = out;                                        // (B,T,V)
    float* hfin = out + (size_t)Bdim * Tdim * Vdim;           // (1,B,H)
    float* attn = hfin + (size_t)Bdim * Hdim;                 // (B,T,S)

    float* ws    = (float*)d_ws;
    float* keys  = ws;                                        // (B,S,H)
    float* hbuf0 = keys + (size_t)Bdim * Sdim * Hdim;         // (B,H)
    float* hbuf1 = hbuf0 + (size_t)Bdim * Hdim;               // (B,H)

    // keys_proj = encoder_outputs @ Ua^T + bu  (one-time WMMA GEMM)
    {
        const int M = Bdim * Sdim, N = Hdim, K = Hdim;
        const int Nt = (N + 15) / 16;
        const int tiles = (M / 16) * Nt;
        const int blocks = (tiles + 7) / 8;
        gemm_wmma_f32<<<blocks, 256, 0, stream>>>(eo, K, Ua, K, bu,
                                                  keys, (long long)N,
                                                  N, K, Nt, tiles);
    }

    // h0 = encoder_hidden[0]
    copy_kernel<<<(Bdim * Hdim + 255) / 256, 256, 0, stream>>>(hbuf0, ehid, Bdim * Hdim);

    const int NT_logits = (Vdim + 15) / 16;                   // 1247
    const int logits_blocks = (NT_logits + 7) / 8;            // 156

    float* hb[2] = { hbuf0, hbuf1 };
    int cur = 0;
    for (int t = 0; t < Tdim; ++t) {
        const float* hin = hb[cur];
        float* hout = hb[1 - cur];

        step_kernel<<<Bdim, 128, 0, stream>>>(eo, keys,
                                              c0s + (size_t)t * Bdim * Hdim,
                                              tgt, emb, Wa, ba, Va, bv,
                                              Wih, bih, Whh, bhh,
                                              hin, hout,
                                              attn + (size_t)t * Sdim, t);

        // logits[b, t, :] = h1 @ W_out^T + b_out  (M-register-blocked WMMA)
        logits_wmma_f32<<<logits_blocks, 256, 0, stream>>>(hout, Wout, bout,
                                                           logp + (size_t)t * Vdim,
                                                           (long long)Tdim * Vdim);

        // in-place log_softmax over V for this step
        logsoftmax_kernel<<<Bdim, 256, 0, stream>>>(logp + (size_t)t * Vdim,
                                                    (long long)Tdim * Vdim);
        cur = 1 - cur;
    }

    // h_final
    copy_kernel<<<(Bdim * Hdim + 255) / 256, 256, 0, stream>>>(hfin, hb[cur], Bdim * Hdim);
}